// NODE_REN_54305566490960
// MI455X (gfx1250) — compile-verified
//
#include <hip/hip_runtime.h>
#include <math.h>

#define NXc 128
#define NYc 64
#define NUc 64
#define NQc 128
#define EPSV 0.01f

typedef __attribute__((ext_vector_type(2))) float v2f;
typedef __attribute__((ext_vector_type(8))) float v8f;

// ---------------------------------------------------------------------------
// WMMA helper: D = A(16x4) * B(4x16) + C, fp32 everywhere.
// 8-arg pattern per probe: (neg_a, A, neg_b, B, c_mod, C, reuse_a, reuse_b)
// ---------------------------------------------------------------------------
__device__ __forceinline__ v8f wmma4(v2f a, v2f b, v8f c) {
  return __builtin_amdgcn_wmma_f32_16x16x4_f32(false, a, false, b, (short)0, c,
                                               false, false);
}

// A-frag (16x4, MxK) from row-major W[ld] at (row0, k0):
//   lanes 0-15 : row = row0+lane,    cols k0,k0+1
//   lanes 16-31: row = row0+lane-16, cols k0+2,k0+3
// B-frag (4x16, KxN) from row-major data Xd (B[K][N] = Xd[row0+N][K]) has the
// identical per-lane addressing, so one loader serves both operands.
__device__ __forceinline__ v2f load_frag(const float* base, int ld, int row0,
                                         int k0, int lane) {
  int r = row0 + (lane & 15);
  int c = k0 + ((lane >> 4) << 1);
  const float* p = base + (long)r * ld + c;
  v2f f;
  f.x = p[0];
  f.y = p[1];
  return f;
}

// ---------------------------------------------------------------------------
// Parameter-update kernels (tiny vs. the batch pass)
// ---------------------------------------------------------------------------
__global__ void k_prep_P(const float* __restrict__ Pstar, float* __restrict__ P) {
  int idx = blockIdx.x * blockDim.x + threadIdx.x;
  if (idx >= NXc * NXc) return;
  int i = idx >> 7, j = idx & 127;
  float s = 0.f;
  for (int k = 0; k < NXc; ++k) s += Pstar[i * NXc + k] * Pstar[j * NXc + k];
  P[idx] = 0.5f * s + (i == j ? EPSV : 0.f);
}

__global__ void k_lam(const float* __restrict__ X, float* __restrict__ lam) {
  int q = threadIdx.x;
  if (q >= NQc) return;
  const float* xr = X + (long)(NXc + q) * (NXc + NQc);
  float s = 0.f;
  for (int k = 0; k < NXc + NQc; ++k) s += xr[k] * xr[k];
  lam[q] = 0.5f * (s + EPSV);
}

// Y = -0.5*(H1 + Y1 - Y1^T)   (ALPHA == 0)
// RHS2 = -H2 - Chi
// D11 = -tril(H4,-1)/lam[:,None]
// C1  = Chi^T / lam[:,None]
__global__ void k_derived(const float* __restrict__ X,
                          const float* __restrict__ Y1,
                          const float* __restrict__ Chi,
                          const float* __restrict__ lam,
                          float* __restrict__ Yo, float* __restrict__ RHS2,
                          float* __restrict__ D11, float* __restrict__ C1) {
  int idx = blockIdx.x * blockDim.x + threadIdx.x;
  if (idx >= NXc * NXc) return;
  int i = idx >> 7, j = idx & 127;
  const int ldx = NXc + NQc;
  const float* Xi = X + (long)i * ldx;
  const float* Xj = X + (long)j * ldx;
  const float* Xhi = X + (long)(NXc + i) * ldx;
  const float* Xhj = X + (long)(NXc + j) * ldx;
  float h1 = 0.f, h2 = 0.f, h4 = 0.f;
  for (int k = 0; k < ldx; ++k) {
    h1 += Xi[k] * Xj[k];
    h2 += Xi[k] * Xhj[k];
    h4 += Xhi[k] * Xhj[k];
  }
  if (i == j) h1 += EPSV;
  Yo[idx] = -0.5f * (h1 + Y1[i * NXc + j] - Y1[j * NXc + i]);
  RHS2[idx] = -h2 - Chi[i * NQc + j];
  D11[idx] = (j < i) ? (-h4 / lam[i]) : 0.f;
  C1[idx] = Chi[j * NQc + i] / lam[i]; // C1[q][k] = Chi[k][q]/lam[q]
}

// Cholesky of P (in LDS) + two 128-RHS triangular solves: A=P^-1 Y, B1=P^-1 RHS2
__global__ __launch_bounds__(256) void k_chol_solve(
    const float* __restrict__ P, const float* __restrict__ Y,
    const float* __restrict__ RHS2, float* __restrict__ A,
    float* __restrict__ B1) {
  __shared__ float L[NXc * NXc]; // 64 KB
  int tid = threadIdx.x;
  for (int t = tid; t < NXc * NXc; t += 256) L[t] = P[t];
  __syncthreads();
  for (int k = 0; k < NXc; ++k) {
    if (tid == 0) L[k * NXc + k] = sqrtf(L[k * NXc + k]);
    __syncthreads();
    float dk = L[k * NXc + k];
    for (int i = k + 1 + tid; i < NXc; i += 256) L[i * NXc + k] /= dk;
    __syncthreads();
    int m = NXc - 1 - k;
    for (int t = tid; t < m * m; t += 256) {
      int i = k + 1 + t / m;
      int j = k + 1 + t % m;
      if (j <= i) L[i * NXc + j] -= L[i * NXc + k] * L[j * NXc + k];
    }
    __syncthreads();
  }
  // 256 threads = 256 RHS columns (128 of Y -> A, 128 of RHS2 -> B1)
  const float* rhs = (tid < NXc) ? Y : RHS2;
  float* out = (tid < NXc) ? A : B1;
  int c = tid & (NXc - 1);
  float z[NXc];
  for (int i = 0; i < NXc; ++i) {
    float s = rhs[i * NXc + c];
    for (int j = 0; j < i; ++j) s -= L[i * NXc + j] * z[j];
    z[i] = s / L[i * NXc + i];
  }
  for (int i = NXc - 1; i >= 0; --i) {
    float s = z[i];
    for (int j = i + 1; j < NXc; ++j) s -= L[j * NXc + i] * z[j];
    z[i] = s / L[i * NXc + i];
    out[i * NXc + c] = z[i];
  }
}

// ---------------------------------------------------------------------------
// Forward pass: 8 waves/block, 16 batch rows per wave, all GEMMs via
// V_WMMA_F32_16X16X4_F32; the strictly-lower-triangular D11 recursion is
// blocked 16-wide (inter-block = WMMA, intra-block = 16 serial tanh steps).
// ---------------------------------------------------------------------------
__global__ __launch_bounds__(256) void k_forward(
    int batch, const float* __restrict__ xi, const float* __restrict__ u,
    const float* __restrict__ Am, const float* __restrict__ B1,
    const float* __restrict__ C1, const float* __restrict__ D11,
    const float* __restrict__ B2, const float* __restrict__ D12,
    const float* __restrict__ C2, const float* __restrict__ D21,
    const float* __restrict__ D22, float* __restrict__ xi_dot,
    float* __restrict__ yi) {
  __shared__ float accT[8][16][17];
  __shared__ float wtile[8][16][17];

  const int lane = threadIdx.x & 31;
  const int wv = threadIdx.x >> 5;
  const long b0 = ((long)blockIdx.x * 8 + wv) * 16;
  if (b0 >= batch) return; // wave-uniform exit; EXEC stays all-1 for WMMA

  const int n = lane & 15;          // batch column within tile / N index
  const int mo = (lane >> 4) << 3;  // +8 row offset for upper half-wave (C/D)
  const int ro = (lane >> 4) << 1;  // +2 K offset for upper half-wave (A/B)

  // Register-resident B-operand fragments (reused across all output tiles)
  v2f xiB[32], uB[16], wB[32];
#pragma unroll
  for (int kk = 0; kk < 32; ++kk) xiB[kk] = load_frag(xi, NXc, (int)b0, kk * 4, lane);
#pragma unroll
  for (int kk = 0; kk < 16; ++kk) uB[kk] = load_frag(u, NUc, (int)b0, kk * 4, lane);

  // ---- blocked tanh recursion over the 8 q-blocks ----
  for (int qb = 0; qb < 8; ++qb) {
    v8f acc = {0.f, 0.f, 0.f, 0.f, 0.f, 0.f, 0.f, 0.f};
    // base = xi @ C1^T + u @ D12^T   (this q-block's 16 rows)
#pragma unroll
    for (int kk = 0; kk < 32; ++kk)
      acc = wmma4(load_frag(C1, NXc, qb * 16, kk * 4, lane), xiB[kk], acc);
#pragma unroll
    for (int kk = 0; kk < 16; ++kk)
      acc = wmma4(load_frag(D12, NUc, qb * 16, kk * 4, lane), uB[kk], acc);
    // + D11[qb, jb<qb] @ w[jb]
    for (int jb = 0; jb < qb; ++jb) {
#pragma unroll
      for (int t = 0; t < 4; ++t)
        acc = wmma4(load_frag(D11, NQc, qb * 16, jb * 16 + t * 4, lane),
                    wB[jb * 4 + t], acc);
    }
    // spill acc tile to LDS (C/D layout -> [row][col])
#pragma unroll
    for (int r = 0; r < 8; ++r) accT[wv][r + mo][n] = acc[r];
    __builtin_amdgcn_wave_barrier(); // same-wave DS ops are in-order

    // intra-block serial tanh on lanes 0-15 (one batch column each);
    // D11 intra-block entries are lane-uniform (scalar loads)
    if (lane < 16) {
      float wloc[16];
#pragma unroll
      for (int i = 0; i < 16; ++i) {
        float v = accT[wv][i][n];
        for (int j = 0; j < i; ++j)
          v += D11[(qb * 16 + i) * NQc + (qb * 16 + j)] * wloc[j];
        wloc[i] = tanhf(v);
        wtile[wv][i][n] = wloc[i];
      }
    }
    __builtin_amdgcn_wave_barrier();
    // rebuild w B-fragments for this block from LDS
#pragma unroll
    for (int t = 0; t < 4; ++t) {
      v2f f;
      f.x = wtile[wv][4 * t + ro + 0][n];
      f.y = wtile[wv][4 * t + ro + 1][n];
      wB[qb * 4 + t] = f;
    }
    __builtin_amdgcn_wave_barrier();
  }

  // ---- xi_dot = xi@A^T + w@B1^T + u@B2^T  (8 feature tiles) ----
  for (int mb = 0; mb < 8; ++mb) {
    v8f acc = {0.f, 0.f, 0.f, 0.f, 0.f, 0.f, 0.f, 0.f};
#pragma unroll
    for (int kk = 0; kk < 32; ++kk)
      acc = wmma4(load_frag(Am, NXc, mb * 16, kk * 4, lane), xiB[kk], acc);
#pragma unroll
    for (int kk = 0; kk < 32; ++kk)
      acc = wmma4(load_frag(B1, NQc, mb * 16, kk * 4, lane), wB[kk], acc);
#pragma unroll
    for (int kk = 0; kk < 16; ++kk)
      acc = wmma4(load_frag(B2, NUc, mb * 16, kk * 4, lane), uB[kk], acc);
#pragma unroll
    for (int r = 0; r < 8; ++r)
      xi_dot[(b0 + n) * NXc + mb * 16 + mo + r] = acc[r];
  }

  // ---- yi = xi@C2^T + w@D21^T + u@D22^T  (4 feature tiles) ----
  for (int mb = 0; mb < 4; ++mb) {
    v8f acc = {0.f, 0.f, 0.f, 0.f, 0.f, 0.f, 0.f, 0.f};
#pragma unroll
    for (int kk = 0; kk < 32; ++kk)
      acc = wmma4(load_frag(C2, NXc, mb * 16, kk * 4, lane), xiB[kk], acc);
#pragma unroll
    for (int kk = 0; kk < 32; ++kk)
      acc = wmma4(load_frag(D21, NQc, mb * 16, kk * 4, lane), wB[kk], acc);
#pragma unroll
    for (int kk = 0; kk < 16; ++kk)
      acc = wmma4(load_frag(D22, NUc, mb * 16, kk * 4, lane), uB[kk], acc);
#pragma unroll
    for (int r = 0; r < 8; ++r)
      yi[(b0 + n) * NYc + mb * 16 + mo + r] = acc[r];
  }
}

// ---------------------------------------------------------------------------
extern "C" void kernel_launch(void* const* d_in, const int* in_sizes, int n_in,
                              void* d_out, int out_size, void* d_ws,
                              size_t ws_size, hipStream_t stream) {
  // inputs: 0=t 1=xi 2=u 3=Pstar 4=Chi 5=Y1 6=X 7=B2 8=D12 9=C2 10=D21 11=D22
  const float* xi = (const float*)d_in[1];
  const float* u = (const float*)d_in[2];
  const float* Pstar = (const float*)d_in[3];
  const float* Chi = (const float*)d_in[4];
  const float* Y1 = (const float*)d_in[5];
  const float* X = (const float*)d_in[6];
  const float* B2 = (const float*)d_in[7];
  const float* D12 = (const float*)d_in[8];
  const float* C2 = (const float*)d_in[9];
  const float* D21 = (const float*)d_in[10];
  const float* D22 = (const float*)d_in[11];

  const int batch = in_sizes[1] / NXc;

  float* ws = (float*)d_ws;
  float* P = ws + 0;
  float* Yo = ws + 16384;
  float* RHS2 = ws + 32768;
  float* D11 = ws + 49152;
  float* C1 = ws + 65536;
  float* Am = ws + 81920;
  float* B1 = ws + 98304;
  float* lam = ws + 114688;

  float* xi_dot = (float*)d_out;
  float* yo = xi_dot + (long)batch * NXc;

  k_prep_P<<<64, 256, 0, stream>>>(Pstar, P);
  k_lam<<<1, 128, 0, stream>>>(X, lam);
  k_derived<<<64, 256, 0, stream>>>(X, Y1, Chi, lam, Yo, RHS2, D11, C1);
  k_chol_solve<<<1, 256, 0, stream>>>(P, Yo, RHS2, Am, B1);

  int nblocks = (batch / 16 + 7) / 8;
  k_forward<<<nblocks, 256, 0, stream>>>(batch, xi, u, Am, B1, C1, D11, B2,
                                         D12, C2, D21, D22, xi_dot, yo);
}